// Tacotron1_72430328479764
// MI455X (gfx1250) — compile-verified
//
#include <hip/hip_runtime.h>
#include <hip/hip_bf16.h>
#include <math.h>

typedef __attribute__((ext_vector_type(2))) float v2f;
typedef __attribute__((ext_vector_type(8))) float v8f;

// Problem dims (compile-time)
#define B_  64
#define S_  512
#define E_  512
#define A_  256
#define HP_ 256
#define P_  128
#define D_  1024
#define M_  128
#define T_  800

// ---------------------------------------------------------------------------
// Generic fp32 WMMA GEMM: C[M x N] = A[M x K] @ W[K x N] + bias
//   block = 256 threads = 8 waves; each wave computes a 16x32 strip
//   (two 16x16 WMMA tiles sharing one A fragment -> 2 wmma per A load)
//   block tile = 64 rows x 64 cols; grid = (N/64, M/64)
//   Uses V_WMMA_F32_16X16X4_F32 (full fp32 precision, CDNA5 matrix pipe).
// A-frag layout (16x4 f32): lanes 0-15 -> K={0,1}, lanes 16-31 -> K={2,3}
// B-frag layout (4x16 f32): vgpr0: rows K0/K2, vgpr1: rows K1/K3 across halves
// C/D layout: vgpr i -> row i (lanes 0-15) / row i+8 (lanes 16-31)
// ---------------------------------------------------------------------------
__global__ void gemm_wmma_f32(const float* __restrict__ A,
                              const float* __restrict__ W,
                              const float* __restrict__ bias,
                              float* __restrict__ C,
                              int K, int lda, int N, int ldc)
{
  const int tid  = threadIdx.x;
  const int wave = tid >> 5;
  const int lane = tid & 31;
  const int half = lane >> 4;    // 0: lanes 0-15, 1: lanes 16-31
  const int l    = lane & 15;
  const int row0 = blockIdx.y * 64 + (wave & 3) * 16;
  const int colg = blockIdx.x * 64 + (wave >> 2) * 32;   // 32-col strip

  v8f c0 = {};
  v8f c1 = {};
  const float* __restrict__ Arow = A + (size_t)(row0 + l) * lda;

  #pragma unroll 4
  for (int k0 = 0; k0 < K; k0 += 4) {
    v2f a, b0, b1;
    a.x = Arow[k0 + 2 * half];
    a.y = Arow[k0 + 2 * half + 1];
    const float* __restrict__ Wr0 = W + (size_t)(k0 + 2 * half) * N;
    const float* __restrict__ Wr1 = Wr0 + N;
    b0.x = Wr0[colg + l];
    b0.y = Wr1[colg + l];
    b1.x = Wr0[colg + 16 + l];
    b1.y = Wr1[colg + 16 + l];
    // prefetch B operand ~16 K-rows ahead (global_prefetch_b8, L2-resident W)
    if ((k0 & 15) == 0 && k0 + 16 < K)
      __builtin_prefetch(W + (size_t)(k0 + 16 + 2 * half) * N + colg + l, 0, 1);
    c0 = __builtin_amdgcn_wmma_f32_16x16x4_f32(
            false, a, false, b0, (short)0, c0, false, false);
    c1 = __builtin_amdgcn_wmma_f32_16x16x4_f32(
            false, a, false, b1, (short)0, c1, false, false);
  }

  const float bv0 = bias ? bias[colg + l]      : 0.0f;
  const float bv1 = bias ? bias[colg + 16 + l] : 0.0f;
  #pragma unroll
  for (int i = 0; i < 8; ++i) {
    C[(size_t)(row0 + i + 8 * half) * ldc + colg + l]      = c0[i] + bv0;
    C[(size_t)(row0 + i + 8 * half) * ldc + colg + 16 + l] = c1[i] + bv1;
  }
}

// ---------------------------------------------------------------------------
// Fused attention (+ prenet in one extra block).
//   blocks 0..63 : attention for batch row b
//     scores[s] = v . tanh(enc_proj[b,s,:] + dec_proj[b,:]) + bv
//     aw = softmax_s(scores);  context = sum_s aw[s] * enc[b,s,:]
//     writes context into x[b,128:640] and y[b,1024:1536], aw into d_out
//   block 64 : prenet for all 64 batch rows (teacher-forced input)
//     pre = relu(relu(prev_mel@W1+b1)@W2+b2)  -> x[b,0:128]
// ---------------------------------------------------------------------------
__global__ void attn_prenet_kernel(const float* __restrict__ enc,
                                   const float* __restrict__ enc_proj,
                                   const float* __restrict__ dec_proj,
                                   const float* __restrict__ v,
                                   const float* __restrict__ bv,
                                   const float* __restrict__ tmels,
                                   const float* __restrict__ W1,
                                   const float* __restrict__ b1,
                                   const float* __restrict__ W2,
                                   const float* __restrict__ b2,
                                   float* __restrict__ x,
                                   float* __restrict__ y,
                                   float* __restrict__ aw_base,
                                   int t)
{
  __shared__ float hid[B_ * HP_];   // prenet stage-1 activations (64 KB)
  __shared__ float dp[A_];
  __shared__ float sc[S_];
  __shared__ float red[16];

  const int tid = threadIdx.x;
  const int b   = blockIdx.x;

  if (b < B_) {
    // ---- attention for batch row b ----
    dp[tid] = dec_proj[b * A_ + tid];            // 256 threads, 256 values
    __syncthreads();

    for (int s = tid; s < S_; s += 256) {
      const float* ep = enc_proj + ((size_t)(b * S_ + s)) * A_;
      float acc = bv[0];
      #pragma unroll 4
      for (int a = 0; a < A_; ++a)
        acc += tanhf(ep[a] + dp[a]) * v[a];
      sc[s] = acc;
    }
    __syncthreads();

    // softmax over S=512 (two elements per thread)
    float m = -3.4e38f;
    for (int s = tid; s < S_; s += 256) m = fmaxf(m, sc[s]);
    for (int off = 16; off; off >>= 1) m = fmaxf(m, __shfl_xor(m, off, 32));
    if ((tid & 31) == 0) red[tid >> 5] = m;
    __syncthreads();
    if (tid == 0) {
      float mm = red[0];
      for (int i = 1; i < 8; ++i) mm = fmaxf(mm, red[i]);
      red[8] = mm;
    }
    __syncthreads();
    m = red[8];

    float esum = 0.0f;
    for (int s = tid; s < S_; s += 256) {
      float e = __expf(sc[s] - m);
      sc[s] = e;
      esum += e;
    }
    for (int off = 16; off; off >>= 1) esum += __shfl_xor(esum, off, 32);
    if ((tid & 31) == 0) red[tid >> 5] = esum;
    __syncthreads();
    if (tid == 0) {
      float ss = 0.0f;
      for (int i = 0; i < 8; ++i) ss += red[i];
      red[9] = 1.0f / ss;
    }
    __syncthreads();
    const float inv = red[9];

    for (int s = tid; s < S_; s += 256) {
      float a_ = sc[s] * inv;
      sc[s] = a_;
      aw_base[(size_t)b * T_ * S_ + (size_t)t * S_ + s] = a_;
    }
    __syncthreads();

    // context: each thread owns 2 output channels (coalesced across lanes)
    const int e0 = tid, e1 = tid + 256;
    float c0 = 0.0f, c1 = 0.0f;
    for (int s = 0; s < S_; ++s) {
      const float* row = enc + ((size_t)(b * S_ + s)) * E_;
      const float a_ = sc[s];
      c0 += a_ * row[e0];
      c1 += a_ * row[e1];
    }
    x[b * (P_ + E_) + P_ + e0] = c0;
    x[b * (P_ + E_) + P_ + e1] = c1;
    y[b * (D_ + E_) + D_ + e0] = c0;
    y[b * (D_ + E_) + D_ + e1] = c1;
  } else {
    // ---- prenet for all batch rows ----
    // stage 1: hid[bb,j] = relu(b1[j] + prev_mel[bb,:] @ W1[:,j])
    for (int idx = tid; idx < B_ * HP_; idx += 256) {
      const int bb = idx >> 8, j = idx & (HP_ - 1);
      float acc = b1[j];
      if (t > 0) {
        const float* pm = tmels + (size_t)bb * T_ * M_ + (size_t)(t - 1) * M_;
        #pragma unroll 4
        for (int mm = 0; mm < M_; ++mm)
          acc += pm[mm] * W1[mm * HP_ + j];
      }
      hid[idx] = fmaxf(acc, 0.0f);
    }
    __syncthreads();
    // stage 2: pre[bb,p] = relu(b2[p] + hid[bb,:] @ W2[:,p]) -> x[bb,0:128]
    for (int idx = tid; idx < B_ * P_; idx += 256) {
      const int bb = idx >> 7, p = idx & (P_ - 1);
      float acc = b2[p];
      const float* hrow = hid + bb * HP_;
      #pragma unroll 4
      for (int j = 0; j < HP_; ++j)
        acc += hrow[j] * W2[j * P_ + p];
      x[bb * (P_ + E_) + p] = fmaxf(acc, 0.0f);
    }
  }
}

// ---------------------------------------------------------------------------
// GRU elementwise update: h_new = (1-z)*n + z*h ; also packs y[:,0:1024]=h_new
// ---------------------------------------------------------------------------
__global__ void gru_update_kernel(const float* __restrict__ gi,
                                  const float* __restrict__ gh,
                                  const float* __restrict__ h_old,
                                  float* __restrict__ h_new,
                                  float* __restrict__ y)
{
  const int idx = blockIdx.x * 256 + threadIdx.x;     // B_*D_ = 65536
  const int b = idx >> 10, c = idx & (D_ - 1);
  const size_t base = (size_t)b * 3 * D_;
  const float ir = gi[base + c],            hr = gh[base + c];
  const float iz = gi[base + D_ + c],       hz = gh[base + D_ + c];
  const float in_ = gi[base + 2 * D_ + c],  hn = gh[base + 2 * D_ + c];
  const float r = 1.0f / (1.0f + __expf(-(ir + hr)));
  const float z = 1.0f / (1.0f + __expf(-(iz + hz)));
  const float n = tanhf(in_ + r * hn);
  const float hv = h_old[idx];
  const float out = (1.0f - z) * n + z * hv;
  h_new[idx] = out;
  y[(size_t)b * (D_ + E_) + c] = out;
}

// ---------------------------------------------------------------------------
// stop token: stop[b] = y[b,:] @ Ws + bs  (tiny, 64 threads)
// ---------------------------------------------------------------------------
__global__ void stop_kernel(const float* __restrict__ y,
                            const float* __restrict__ Ws,
                            const float* __restrict__ bs,
                            float* __restrict__ stop_base,
                            int t)
{
  const int b = threadIdx.x;
  if (b >= B_) return;
  float acc = bs[0];
  const float* yr = y + (size_t)b * (D_ + E_);
  #pragma unroll 4
  for (int k = 0; k < D_ + E_; ++k) acc += yr[k] * Ws[k];
  stop_base[(size_t)b * T_ + t] = acc;
}

__global__ void zero_kernel(float* __restrict__ p, int n)
{
  const int i = blockIdx.x * 256 + threadIdx.x;
  if (i < n) p[i] = 0.0f;
}

// ---------------------------------------------------------------------------
extern "C" void kernel_launch(void* const* d_in, const int* in_sizes, int n_in,
                              void* d_out, int out_size, void* d_ws, size_t ws_size,
                              hipStream_t stream)
{
  const float* enc    = (const float*)d_in[0];   // (B,S,E)
  const float* tmels  = (const float*)d_in[1];   // (B,T,M)
  const float* We     = (const float*)d_in[2];   // (E,A)
  const float* be     = (const float*)d_in[3];
  const float* Wd     = (const float*)d_in[4];   // (D,A)
  const float* bd     = (const float*)d_in[5];
  const float* vvec   = (const float*)d_in[6];   // (A,)
  const float* bv     = (const float*)d_in[7];
  const float* W1     = (const float*)d_in[8];   // (M,Hp)
  const float* b1     = (const float*)d_in[9];
  const float* W2     = (const float*)d_in[10];  // (Hp,P)
  const float* b2     = (const float*)d_in[11];
  const float* W_ih   = (const float*)d_in[12];  // (P+E, 3D)
  const float* b_ih   = (const float*)d_in[13];
  const float* W_hh   = (const float*)d_in[14];  // (D, 3D)
  const float* b_hh   = (const float*)d_in[15];
  const float* Wo     = (const float*)d_in[16];  // (D+E, M)
  const float* bo     = (const float*)d_in[17];
  const float* Ws     = (const float*)d_in[18];  // (D+E,)
  const float* bs     = (const float*)d_in[19];

  float* out_mel  = (float*)d_out;                                   // (B,T,M)
  float* out_stop = out_mel + (size_t)B_ * T_ * M_;                  // (B,T)
  float* out_aw   = out_stop + (size_t)B_ * T_;                      // (B,T,S)

  // Workspace layout (floats)
  float* ws       = (float*)d_ws;
  float* enc_proj = ws;                                  // B*S*A = 8,388,608
  float* h0       = enc_proj + (size_t)B_ * S_ * A_;     // 65,536
  float* h1       = h0 + (size_t)B_ * D_;                // 65,536
  float* xbuf     = h1 + (size_t)B_ * D_;                // B*(P+E) = 40,960
  float* ybuf     = xbuf + (size_t)B_ * (P_ + E_);       // B*(D+E) = 98,304
  float* gi       = ybuf + (size_t)B_ * (D_ + E_);       // B*3D = 196,608
  float* gh       = gi + (size_t)B_ * 3 * D_;            // 196,608
  float* dec_proj = gh + (size_t)B_ * 3 * D_;            // B*A = 16,384

  // h(0) = 0
  zero_kernel<<<dim3((B_ * D_ + 255) / 256), dim3(256), 0, stream>>>(h0, B_ * D_);

  // enc_proj = enc.reshape(B*S, E) @ We + be   (one-time, WMMA)
  gemm_wmma_f32<<<dim3(A_ / 64, (B_ * S_) / 64), dim3(256), 0, stream>>>(
      enc, We, be, enc_proj, /*K=*/E_, /*lda=*/E_, /*N=*/A_, /*ldc=*/A_);

  for (int t = 0; t < T_; ++t) {
    float* hc = (t & 1) ? h1 : h0;
    float* hn = (t & 1) ? h0 : h1;

    // dec_proj = h @ Wd + bd                       (64x1024x256, WMMA)
    gemm_wmma_f32<<<dim3(A_ / 64, 1), dim3(256), 0, stream>>>(
        hc, Wd, bd, dec_proj, D_, D_, A_, A_);

    // attention + softmax + context (+ prenet block)
    attn_prenet_kernel<<<dim3(B_ + 1), dim3(256), 0, stream>>>(
        enc, enc_proj, dec_proj, vvec, bv, tmels,
        W1, b1, W2, b2, xbuf, ybuf, out_aw, t);

    // gi = [pre, ctx] @ W_ih + b_ih               (64x640x3072, WMMA)
    gemm_wmma_f32<<<dim3(3 * D_ / 64, 1), dim3(256), 0, stream>>>(
        xbuf, W_ih, b_ih, gi, P_ + E_, P_ + E_, 3 * D_, 3 * D_);

    // gh = h @ W_hh + b_hh                        (64x1024x3072, WMMA)
    gemm_wmma_f32<<<dim3(3 * D_ / 64, 1), dim3(256), 0, stream>>>(
        hc, W_hh, b_hh, gh, D_, D_, 3 * D_, 3 * D_);

    // GRU elementwise update; packs y = [h_new, ctx]
    gru_update_kernel<<<dim3(B_ * D_ / 256), dim3(256), 0, stream>>>(
        gi, gh, hc, hn, ybuf);

    // mel = y @ Wo + bo -> d_out[:, t, :]         (64x1536x128, WMMA)
    gemm_wmma_f32<<<dim3(M_ / 64, 1), dim3(256), 0, stream>>>(
        ybuf, Wo, bo, out_mel + (size_t)t * M_, D_ + E_, D_ + E_, M_, T_ * M_);

    // stop[b, t]
    stop_kernel<<<dim3(1), dim3(64), 0, stream>>>(ybuf, Ws, bs, out_stop, t);
  }
}